// Net_16329465660089
// MI455X (gfx1250) — compile-verified
//
#include <hip/hip_runtime.h>
#include <hip/hip_bf16.h>

// ---------------------------------------------------------------------------
// Problem constants (match reference)
// ---------------------------------------------------------------------------
#define N_NODES 10000
#define N_EDGES 4096
#define IN_F    300
#define OUT_F   150
#define NPAD    160                      // OUT_F padded to 16-col tiles

// Tiling: one wave owns a 16(edges) x 32(out-cols) tile -> 4 f32 accumulators.
#define MT 16
#define NTILES_M   (N_EDGES / MT)        // 256
#define NPAIRS_N   (NPAD / 32)           // 5  (pairs of 16-col tiles)
#define NTILES_TOT (NTILES_M * NPAIRS_N) // 1280 wave-tiles
#define WAVES_PER_BLOCK 4

typedef __attribute__((ext_vector_type(2))) float v2f;
typedef __attribute__((ext_vector_type(4))) float v4f;
typedef __attribute__((ext_vector_type(8))) float v8f;

#define WBP_ELEMS ((IN_F / 2) * NPAD)    // 150*160 float4 = 384 KB

// ---------------------------------------------------------------------------
// Zero sums + counts
// ---------------------------------------------------------------------------
__global__ void gnn_zero_ws(float* __restrict__ p, int n) {
    int i = blockIdx.x * blockDim.x + threadIdx.x;
    if (i < n) p[i] = 0.0f;
}

// ---------------------------------------------------------------------------
// Pack W_edge / b_edge into fragment-ordered, zero-padded float4 table:
//   WBp[k2*NPAD + n] = { W[2k2, n], W[2k2+1, n], b[2k2, n], b[2k2+1, n] }
// (n >= OUT_F -> zeros). One b128 load then feeds both accumulators.
// ---------------------------------------------------------------------------
__global__ void gnn_pack_weights(const float* __restrict__ W,
                                 const float* __restrict__ Bm,
                                 v4f* __restrict__ WBp) {
    int i = blockIdx.x * blockDim.x + threadIdx.x;
    if (i >= WBP_ELEMS) return;
    const int k2 = i / NPAD;
    const int n  = i % NPAD;
    v4f v = {0.0f, 0.0f, 0.0f, 0.0f};
    if (n < OUT_F) {
        v.x = W [(2 * k2 + 0) * OUT_F + n];
        v.y = W [(2 * k2 + 1) * OUT_F + n];
        v.z = Bm[(2 * k2 + 0) * OUT_F + n];
        v.w = Bm[(2 * k2 + 1) * OUT_F + n];
    }
    WBp[i] = v;
}

// ---------------------------------------------------------------------------
// In-degree counts via float atomics
// ---------------------------------------------------------------------------
__global__ void gnn_count_edges(const int* __restrict__ dst, float* __restrict__ cnt) {
    int e = blockIdx.x * blockDim.x + threadIdx.x;
    if (e < N_EDGES) unsafeAtomicAdd(&cnt[dst[e]], 1.0f);
}

// ---------------------------------------------------------------------------
// Fused gather + dual GEMM (WMMA f32 16x16x4) + edge combine + scatter-add.
//
//   accA = features[src] @ W ;  accB = features[src] @ b
//   msg  = efeat[e]*accA + accB ;  sums[dst[e], :] += msg (atomic f32)
//
// Fragments (V_WMMA_F32_16X16X4_F32):
//   A 16x4 : M = lane&15; lanes 0-15 hold K={k0,k0+1}, lanes 16-31 K={k0+2,k0+3}
//   B 4x16 : N = lane&15; mirrored K split               (fed from WBp, b128)
//   C 16x16: VGPR r -> M = r + 8*(lane>=16), N = lane&15
// Inner loop: 1x global_load_b64 (A) + 2x global_load_b128 (WB) + 4x v_wmma.
// No EXEC manipulation inside the loop (pad columns are zero).
// ---------------------------------------------------------------------------
__global__ __launch_bounds__(WAVES_PER_BLOCK * 32)
void gnn_edge_gemm_scatter(const float* __restrict__ feat,   // [N_NODES, IN_F]
                           const float* __restrict__ efeat,  // [N_EDGES]
                           const v4f*   __restrict__ WBp,    // [IN_F/2, NPAD]
                           const int*   __restrict__ src,    // [N_EDGES]
                           const int*   __restrict__ dst,    // [N_EDGES]
                           float*       __restrict__ sums)   // [N_NODES, OUT_F]
{
    const int wave = threadIdx.x >> 5;
    const int lane = threadIdx.x & 31;
    const int tile = blockIdx.x * WAVES_PER_BLOCK + wave;
    if (tile >= NTILES_TOT) return;           // wave-uniform: EXEC stays all-1s

    const int mtile = tile / NPAIRS_N;
    const int npair = tile % NPAIRS_N;
    const int n0    = npair * 32;             // first of 32 columns this wave owns

    const int m     = lane & 15;              // A-row / B-col within tile
    const int khalf = lane >> 4;              // 0: K={0,1}, 1: K={2,3}

    const float* __restrict__ arow = feat + src[mtile * MT + m] * IN_F;
    const v4f*   __restrict__ wb   = WBp + khalf * NPAD + n0 + m;

    v8f accA0 = {}, accB0 = {};
    v8f accA1 = {}, accB1 = {};

    for (int k0 = 0; k0 < IN_F; k0 += 4) {
        v2f a;
        a.x = arow[k0 + 2 * khalf + 0];
        a.y = arow[k0 + 2 * khalf + 1];

        const v4f* wbrow = wb + (k0 >> 1) * NPAD;
        const v4f  wb0   = wbrow[0];          // cols n0 + m
        const v4f  wb1   = wbrow[16];         // cols n0 + 16 + m

        v2f bw0 = {wb0.x, wb0.y}, bb0 = {wb0.z, wb0.w};
        v2f bw1 = {wb1.x, wb1.y}, bb1 = {wb1.z, wb1.w};

        accA0 = __builtin_amdgcn_wmma_f32_16x16x4_f32(
            false, a, false, bw0, (short)0, accA0, false, false);
        accB0 = __builtin_amdgcn_wmma_f32_16x16x4_f32(
            false, a, false, bb0, (short)0, accB0, false, false);
        accA1 = __builtin_amdgcn_wmma_f32_16x16x4_f32(
            false, a, false, bw1, (short)0, accA1, false, false);
        accB1 = __builtin_amdgcn_wmma_f32_16x16x4_f32(
            false, a, false, bb1, (short)0, accB1, false, false);
    }

    // Epilogue: combine with efeat and scatter-add (guards only here).
    const int mhi = khalf * 8;
    const int c0  = n0 + m;                   // tile-0 column
    const int c1  = c0 + 16;                  // tile-1 column
#pragma unroll
    for (int r = 0; r < 8; ++r) {
        const int   e  = mtile * MT + mhi + r;
        const float ef = efeat[e];
        const int   db = dst[e] * OUT_F;
        if (c0 < OUT_F)
            unsafeAtomicAdd(&sums[db + c0], ef * accA0[r] + accB0[r]);
        if (c1 < OUT_F)
            unsafeAtomicAdd(&sums[db + c1], ef * accA1[r] + accB1[r]);
    }
}

// ---------------------------------------------------------------------------
// mean + bias + ReLU
// ---------------------------------------------------------------------------
__global__ void gnn_finalize(const float* __restrict__ sums,
                             const float* __restrict__ cnt,
                             const float* __restrict__ bias,
                             float* __restrict__ out)
{
    int i = blockIdx.x * blockDim.x + threadIdx.x;
    if (i >= N_NODES * OUT_F) return;
    const int node = i / OUT_F;
    const int o    = i % OUT_F;
    float c = cnt[node];
    c = (c < 1.0f) ? 1.0f : c;
    float v = sums[i] / c + bias[o];
    out[i] = (v > 0.0f) ? v : 0.0f;
}

// ---------------------------------------------------------------------------
// kernel_launch
//   d_in order: features, efeat, W_edge, b_edge, bias, src, dst
//   ws layout : WBp (float4[24000], 16B-aligned at base) | sums | cnt
// ---------------------------------------------------------------------------
extern "C" void kernel_launch(void* const* d_in, const int* in_sizes, int n_in,
                              void* d_out, int out_size, void* d_ws, size_t ws_size,
                              hipStream_t stream) {
    const float* feat  = (const float*)d_in[0];
    const float* efeat = (const float*)d_in[1];
    const float* W     = (const float*)d_in[2];   // [IN_F, OUT_F] row-major
    const float* Bm    = (const float*)d_in[3];   // [IN_F, OUT_F] row-major
    const float* bias  = (const float*)d_in[4];
    const int*   src   = (const int*)d_in[5];
    const int*   dst   = (const int*)d_in[6];
    float*       out   = (float*)d_out;

    v4f*   WBp  = (v4f*)d_ws;                          // 384 KB, 16B aligned
    float* sums = (float*)d_ws + 4 * WBP_ELEMS;        // N_NODES*OUT_F floats
    float* cnt  = sums + N_NODES * OUT_F;              // N_NODES floats

    // 1) zero accumulators (every call: graph replays must be deterministic)
    {
        const int n = N_NODES * OUT_F + N_NODES;
        gnn_zero_ws<<<(n + 255) / 256, 256, 0, stream>>>(sums, n);
    }

    // 2) pack weights into fragment-ordered padded float4 table
    gnn_pack_weights<<<(WBP_ELEMS + 255) / 256, 256, 0, stream>>>(W, Bm, WBp);

    // 3) in-degree counts
    gnn_count_edges<<<(N_EDGES + 255) / 256, 256, 0, stream>>>(dst, cnt);

    // 4) fused gather + WMMA dual-GEMM + combine + scatter-add
    {
        const int blocks = (NTILES_TOT + WAVES_PER_BLOCK - 1) / WAVES_PER_BLOCK; // 320
        gnn_edge_gemm_scatter<<<blocks, WAVES_PER_BLOCK * 32, 0, stream>>>(
            feat, efeat, WBp, src, dst, sums);
    }

    // 5) mean + bias + relu
    {
        const int n = N_NODES * OUT_F;
        gnn_finalize<<<(n + 255) / 256, 256, 0, stream>>>(sums, cnt, bias, out);
    }
}